// AdaConv2d_old_53326313947651
// MI455X (gfx1250) — compile-verified
//
#include <hip/hip_runtime.h>

typedef __attribute__((ext_vector_type(2))) float v2f;
typedef __attribute__((ext_vector_type(8))) float v8f;
typedef __attribute__((ext_vector_type(4))) unsigned int v4u;
typedef __attribute__((ext_vector_type(8))) int v8i;
typedef __attribute__((ext_vector_type(4))) int v4i;

#define HH    112
#define WW    112
#define ICN   64
#define OCN   64
#define NDEM  4
#define KTOT  576          // IC * 3 * 3
#define KCH   144          // K-chunk staged in LDS (16 ic * 9 taps)
#define NCH   64           // output pixels per block (4 N-tiles)
#define ASTR  162          // padded A row stride in dwords: 144 + 144/8 pads (TDM pad cfg)
#define BPAD  65           // B LDS stride
#define FUSE_EPOCH 9
#define FUSED_ELEMS (NDEM * OCN * KTOT)   // 147456 floats

#if defined(__has_builtin)
#if __has_builtin(__builtin_amdgcn_tensor_load_to_lds)
#define HAVE_TDM 1
#endif
#endif
#ifndef HAVE_TDM
#define HAVE_TDM 0
#endif

// element (m, kk) of the padded A tile, in dwords
__device__ __forceinline__ int aidx(int m, int kk) {
    return m * ASTR + kk + (kk >> 3);
}

// ---- pre-pass: fused[d][oc][k] = base[oc][k] * mask[d][k] ----
__global__ void fuse_weights(const float* __restrict__ kbase,
                             const float* __restrict__ kmask,
                             float* __restrict__ fused)
{
    int i = blockIdx.x * 256 + threadIdx.x;
    if (i < FUSED_ELEMS) {
        int d   = i / (OCN * KTOT);
        int rem = i - d * (OCN * KTOT);
        int k   = rem % KTOT;
        fused[i] = kbase[rem] * kmask[d * KTOT + k];
    }
}

template <bool USE_WS>
__global__ __launch_bounds__(256, 2)
void adaconv_wmma_f32(const float* __restrict__ x,
                      const float* __restrict__ kbase,
                      const float* __restrict__ kmask,
                      const float* __restrict__ fused,
                      const int*   __restrict__ label,
                      const int*   __restrict__ epoch,
                      float*       __restrict__ out)
{
    __shared__ __align__(16) float Abuf[OCN * ASTR];   // padded 64 x 144 tile
    __shared__ __align__(16) float Bbuf[KCH][BPAD];    // 144 k x 64 px (+pad)

    const int b    = blockIdx.y;
    const int p0   = blockIdx.x * NCH;
    const int t    = threadIdx.x;
    const int lane = t & 31;
    const int wv   = t >> 5;             // 8 waves
    const int n0   = (wv & 3) << 4;      // 4 N-tiles of 16
    const int m0   = (wv >> 2) << 5;     // 2 M-pairs: rows m0 and m0+16
    const int wvs  = __builtin_amdgcn_readfirstlane(t) >> 5;  // scalar wave id

    int d = (epoch[0] >= FUSE_EPOCH) ? 0 : label[b];
    d &= 3;

    const float* xb    = x + (size_t)b * ICN * HH * WW;
    const float* maskd = kmask + (size_t)d * KTOT;
    const float* Aw    = fused + (size_t)d * OCN * KTOT;   // [oc][576]

    v8f acc0 = {};
    v8f acc1 = {};

    for (int kc = 0; kc < KTOT; kc += KCH) {
        // ---- stage A chunk into LDS ----
        if (USE_WS) {
#if HAVE_TDM
            if (wvs == 0) {
                // Tensor Data Mover: DMA the 64x144 f32 tile (of the 64x576
                // fused matrix) into LDS with 1-dword pad every 8 dwords.
                unsigned lds_base = (unsigned)(size_t)&Abuf[0];
                unsigned long long ga = (unsigned long long)(size_t)(Aw + kc);
                v4u g0;
                g0[0] = 1u;                                   // count=1
                g0[1] = lds_base;                             // lds_addr
                g0[2] = (unsigned)ga;                         // global_addr[31:0]
                g0[3] = (unsigned)((ga >> 32) & 0x1FFFFFFu)   // global_addr[56:32]
                        | (2u << 30);                         // type=2 (image)
                v8i g1;
                g1[0] = (2 << 16)      // data_size = 4B
                      | (1 << 20)      // pad_enable
                      | (2 << 22)      // pad_interval: 8 dwords
                      | (0 << 25);     // pad_amount: 1 dword
                g1[1] = (int)(576u << 16);   // tensor_dim0 = 576
                g1[2] = (int)(64u  << 16);   // tensor_dim1 = 64
                g1[3] = (int)(144u << 16);   // tile_dim0   = 144
                g1[4] = 64;                  // tile_dim1   = 64, tile_dim2 = 0
                g1[5] = 576;                 // tensor_dim0_stride = 576
                g1[6] = 0;
                g1[7] = 0;
                v4i gz = {0, 0, 0, 0};       // 2D tile: groups 2/3 zero
#if __clang_major__ >= 23
                v8i gz8 = {0, 0, 0, 0, 0, 0, 0, 0};
                __builtin_amdgcn_tensor_load_to_lds(g0, g1, gz, gz, gz8, 0);
#else
                __builtin_amdgcn_tensor_load_to_lds(g0, g1, gz, gz, 0);
#endif
            }
#else
            // fallback: per-lane async copy (ASYNCcnt path), padded layout
            for (int i = t; i < OCN * KCH; i += 256) {
                int m  = i / KCH;
                int kk = i - m * KCH;
                const float* gp = Aw + (size_t)m * KTOT + kc + kk;
                unsigned ldsa = (unsigned)(size_t)&Abuf[aidx(m, kk)];
                asm volatile("global_load_async_to_lds_b32 %0, %1, off"
                             :: "v"(ldsa), "v"(gp) : "memory");
            }
#endif
        } else {
            // no workspace: fuse in-place into padded layout
            for (int i = t; i < OCN * KCH; i += 256) {
                int m  = i / KCH;
                int kk = i - m * KCH;
                Abuf[aidx(m, kk)] = kbase[m * KTOT + kc + kk] * maskd[kc + kk];
            }
        }

        // ---- stage B: im2col chunk (zero-padded 3x3 window gather),
        //      overlaps the A-tile DMA ----
        for (int i = t; i < KCH * NCH; i += 256) {
            int kk = i >> 6;             // / 64
            int n  = i & 63;
            int k  = kc + kk;
            int ic = k / 9;
            int r  = k - ic * 9;
            int p  = p0 + n;
            int oh = p / WW;
            int ow = p - oh * WW;
            int ih = oh + (r / 3) - 1;
            int iw = ow + (r % 3) - 1;
            float v = 0.0f;
            if ((unsigned)ih < (unsigned)HH && (unsigned)iw < (unsigned)WW)
                v = xb[(ic * HH + ih) * WW + iw];
            Bbuf[kk][n] = v;
        }

        if (USE_WS) {
#if HAVE_TDM
            if (wvs == 0)
                __builtin_amdgcn_s_wait_tensorcnt(0);
#else
            asm volatile("s_wait_asynccnt 0x0" ::: "memory");
#endif
        }
        __syncthreads();

        // ---- 36 K-steps: 1 B fragment reused by 2 WMMA (2 M-tiles) ----
        const int mrow0 = m0 + (lane & 15);
        const int mrow1 = mrow0 + 16;
        const int ncol  = n0 + (lane & 15);
        const int khalf = (lane >> 4) << 1;   // lanes 0-15 -> K{0,1}; 16-31 -> K{2,3}
        #pragma unroll
        for (int kk = 0; kk < KCH; kk += 4) {
            const int kh2 = kk + khalf;
            const int ia0 = aidx(mrow0, kh2);
            const int ia1 = aidx(mrow1, kh2);
            v2f bf, a0, a1;
            bf.x = Bbuf[kh2][ncol];
            bf.y = Bbuf[kh2 + 1][ncol];
            a0.x = Abuf[ia0];
            a0.y = Abuf[ia0 + 1];
            a1.x = Abuf[ia1];
            a1.y = Abuf[ia1 + 1];
            acc0 = __builtin_amdgcn_wmma_f32_16x16x4_f32(
                false, a0, false, bf, (short)0, acc0, false, false);
            acc1 = __builtin_amdgcn_wmma_f32_16x16x4_f32(
                false, a1, false, bf, (short)0, acc1, false, false);
        }
        __syncthreads();
    }

    // ---- write two 16x16 f32 C/D tiles per ISA layout ----
    const int col = lane & 15;
    const int p   = p0 + n0 + col;
    const int mhi = (lane >> 4) << 3;    // VGPR v: M = v (lanes 0-15), v+8 (lanes 16-31)
    float* ob = out + (size_t)b * OCN * (HH * WW);
    #pragma unroll
    for (int v = 0; v < 8; ++v) {
        int oc = m0 + mhi + v;
        ob[(size_t)oc * (HH * WW) + p] = acc0[v];
        ob[(size_t)(oc + 16) * (HH * WW) + p] = acc1[v];
    }
}

extern "C" void kernel_launch(void* const* d_in, const int* in_sizes, int n_in,
                              void* d_out, int out_size, void* d_ws, size_t ws_size,
                              hipStream_t stream) {
    const float* x     = (const float*)d_in[0];
    const float* kbase = (const float*)d_in[1];
    const float* kmask = (const float*)d_in[2];
    const int*   label = (const int*)d_in[3];
    const int*   epoch = (const int*)d_in[4];
    float*       out   = (float*)d_out;

    dim3 grid((HH * WW) / NCH, 64);      // 196 x 64 blocks

    if (ws_size >= (size_t)FUSED_ELEMS * sizeof(float)) {
        float* fused = (float*)d_ws;
        fuse_weights<<<(FUSED_ELEMS + 255) / 256, 256, 0, stream>>>(kbase, kmask, fused);
        adaconv_wmma_f32<true><<<grid, 256, 0, stream>>>(
            x, kbase, kmask, fused, label, epoch, out);
    } else {
        adaconv_wmma_f32<false><<<grid, 256, 0, stream>>>(
            x, kbase, kmask, (const float*)nullptr, label, epoch, out);
    }
}